// FixedQFMClassifier_17171279249726
// MI455X (gfx1250) — compile-verified
//
#include <hip/hip_runtime.h>
#include <math.h>

typedef __attribute__((ext_vector_type(2))) float v2f;
typedef __attribute__((ext_vector_type(8))) float v8f;

#define NQ 14
#define IN_DIM 784
#define NFEAT 105
#define NCLS 10
#define BATCH 2048
#define WAVES_PER_BLOCK 2
#define BLOCK (WAVES_PER_BLOCK * 32)

__global__ __launch_bounds__(BLOCK)
void qfm_fused_kernel(const float* __restrict__ x,
                      const float* __restrict__ proj_w,
                      const float* __restrict__ proj_b,
                      const float* __restrict__ cls_w,
                      const float* __restrict__ cls_b,
                      float* __restrict__ out)
{
    __shared__ float s_clsw[NCLS * NFEAT];
    __shared__ float s_clsb[NCLS];
    __shared__ float s_ctile[WAVES_PER_BLOCK][16][16];

    const int tid = threadIdx.x;
    for (int i = tid; i < NCLS * NFEAT; i += BLOCK) s_clsw[i] = cls_w[i];
    if (tid < NCLS) s_clsb[tid] = cls_b[tid];

    const int wid     = tid >> 5;
    const int lane    = tid & 31;
    const int halfsel = lane >> 4;   // 0: lanes 0-15, 1: lanes 16-31
    const int l16     = lane & 15;
    const int m0      = blockIdx.x * (16 * WAVES_PER_BLOCK) + wid * 16;

    // ---------------- projection GEMM via v_wmma_f32_16x16x4_f32 ----------------
    // C[m,n] = sum_k x[m,k] * proj_w[n,k];  M-tile = 16 batch rows, N = 14 (pad 16)
    // A 16x4 f32 layout: lanes 0-15 hold K={0,1}, lanes 16-31 hold K={2,3}.
    // B columns n>=14: clamp to row 13 (garbage goes to C cols 14/15, never read)
    // -> no predication / EXEC churn inside the K loop.
    const int  wn   = (l16 < NQ) ? l16 : (NQ - 1);
    const float* xrow = x      + (size_t)(m0 + l16) * IN_DIM + 2 * halfsel;
    const float* wrow = proj_w + (size_t)wn         * IN_DIM + 2 * halfsel;

    v8f acc = {};
#pragma unroll 4
    for (int k = 0; k < IN_DIM; k += 4) {
        v2f a, b;
        { const float2 t = *(const float2*)(xrow + k); a[0] = t.x; a[1] = t.y; }
        { const float2 t = *(const float2*)(wrow + k); b[0] = t.x; b[1] = t.y; }
        acc = __builtin_amdgcn_wmma_f32_16x16x4_f32(
                  false, a, false, b, (short)0, acc, false, false);
    }

    // c_q = cos(theta) = cos(pi * tanh(gemm + bias))  (cos^2(t/2)-sin^2(t/2)=cos t)
    const float bias = (l16 < NQ) ? proj_b[l16] : 0.0f;
#pragma unroll
    for (int v = 0; v < 8; ++v) {
        const float t = tanhf(acc[v] + bias);
        s_ctile[wid][v + 8 * halfsel][l16] =
            cosf(3.14159265358979323846f * t);
    }
    __syncthreads();

    // ------- features + classifier: all 32 lanes active -------
    // lane handles row l16; lanes 0-15 -> classes 0-4, lanes 16-31 -> classes 5-9
    const int cls0 = halfsel * 5;

    float c[NQ];
#pragma unroll
    for (int q = 0; q < NQ; ++q) c[q] = s_ctile[wid][l16][q];

    // parity prefix products: P_k = prod_{q<=k, q==k mod 2} c_q
    float P[NQ];
#pragma unroll
    for (int q = 0; q < NQ; ++q) P[q] = (q >= 2) ? c[q] * P[q - 2] : c[q];

    float o[5];
#pragma unroll
    for (int cl = 0; cl < 5; ++cl) o[cl] = s_clsb[cls0 + cl];

    int fidx = 0;
    // single-Z features: <Z_i> = P_i
#pragma unroll
    for (int i = 0; i < NQ; ++i) {
        const float f = P[i];
#pragma unroll
        for (int cl = 0; cl < 5; ++cl)
            o[cl] = fmaf(s_clsw[(cls0 + cl) * NFEAT + fidx], f, o[cl]);
        ++fidx;
    }
    // ZZ features: same parity -> prod_{i<q<=j, q==j mod2} c_q ; opposite -> P_i*P_j
#pragma unroll
    for (int i = 0; i < NQ; ++i) {
        float sp = 1.0f;
#pragma unroll
        for (int j = i + 1; j < NQ; ++j) {
            float f;
            if (((j - i) & 1) == 0) { sp *= c[j]; f = sp; }
            else                    { f = P[i] * P[j]; }
#pragma unroll
            for (int cl = 0; cl < 5; ++cl)
                o[cl] = fmaf(s_clsw[(cls0 + cl) * NFEAT + fidx], f, o[cl]);
            ++fidx;
        }
    }

    float* orow = out + (size_t)(m0 + l16) * NCLS + cls0;
#pragma unroll
    for (int cl = 0; cl < 5; ++cl) orow[cl] = o[cl];
}

extern "C" void kernel_launch(void* const* d_in, const int* in_sizes, int n_in,
                              void* d_out, int out_size, void* d_ws, size_t ws_size,
                              hipStream_t stream) {
    (void)in_sizes; (void)n_in; (void)out_size; (void)d_ws; (void)ws_size;
    const float* x      = (const float*)d_in[0];
    const float* proj_w = (const float*)d_in[1];
    const float* proj_b = (const float*)d_in[2];
    const float* cls_w  = (const float*)d_in[3];
    const float* cls_b  = (const float*)d_in[4];
    float* out = (float*)d_out;

    dim3 grid(BATCH / (16 * WAVES_PER_BLOCK));   // 64 blocks x 2 waves = 128 row-tiles
    dim3 block(BLOCK);
    qfm_fused_kernel<<<grid, block, 0, stream>>>(x, proj_w, proj_b, cls_w, cls_b, out);
}